// Camera_back_projection_layer_42417097016821
// MI455X (gfx1250) — compile-verified
//
#include <hip/hip_runtime.h>
#include <stdint.h>

// ---------------------------------------------------------------------------
// Camera back-projection -> TDF voxel grid, MI455X (gfx1250, wave32).
//
// Phase 1: clear 16*128^3 packed u64 accumulators in d_ws   (b128 stores)
// Phase 2: per-pixel back-project + ONE global_atomic_add_u64 per valid pixel
//          packed accumulator: bits[63:42] = count, bits[41:0] = sum(dist)*2^30
//          (integer packing => deterministic result, half the atomic traffic,
//           accumulator array ~70% resident in 192 MB L2)
// Phase 3: finalize: out = cnt ? 1 - 128*(sum/cnt) : 0  (NT b128 loads, b128 store)
// ---------------------------------------------------------------------------

#define RES   128
#define IMG_W 480
#define IMG_H 480

static constexpr int      NPIX      = IMG_W * IMG_H;               // 230400 (divisible by 256)
static constexpr uint64_t CNT_ONE   = 1ull << 42;                  // count increment
static constexpr uint64_t SUM_MASK  = (1ull << 42) - 1ull;         // sum field
static constexpr float    SUM_SCALE = 1073741824.0f;               // 2^30
static constexpr float    INV_SUM_SCALE = 9.3132257461547852e-10f; // 2^-30
static constexpr float    TRUNC     = 1.0f / 128.0f;

typedef unsigned long long u64x2 __attribute__((ext_vector_type(2)));

// ---- Phase 1: clear accumulators (2 x global_store_b128 per thread) -------
__global__ void __launch_bounds__(256)
clear_acc_kernel(u64x2* __restrict__ acc2) {
    size_t i = ((size_t)blockIdx.x * blockDim.x + threadIdx.x) * 2;
    u64x2 z = {0ull, 0ull};
    acc2[i + 0] = z;
    acc2[i + 1] = z;
}

// ---- Phase 2: back-project + packed u64 scatter-atomic --------------------
__global__ void __launch_bounds__(256)
backproject_kernel(const float* __restrict__ depth,
                   const float* __restrict__ fl,
                   const float* __restrict__ cam_dist,
                   unsigned long long* __restrict__ acc) {
    const int b   = blockIdx.y;                                  // image index
    const int pix = blockIdx.x * blockDim.x + threadIdx.x;       // [0, 480*480)
    if (pix >= NPIX) return;

    const int h = pix / IMG_W;
    const int w = pix - h * IMG_W;

    // Uniform per image -> scalar loads
    const float f     = fl[b];
    const float cd    = cam_dist[b];
    const float inv_f = 1.0f / f;

    const float d = depth[(size_t)b * NPIX + pix];

    const float u = (float)w + 0.5f - (0.5f * IMG_W);
    const float v = (float)h + 0.5f - (0.5f * IMG_H);

    const float x = d * u * inv_f;
    const float y = d * v * inv_f;
    const float z = cd - d;

    const int vx = (int)floorf((x + 0.5f) * (float)RES);
    const int vy = (int)floorf((y + 0.5f) * (float)RES);
    const int vz = (int)floorf((z + 0.5f) * (float)RES);

    const bool valid = (d > 0.0f) &&
                       (vx >= 0) && (vx < RES) &&
                       (vy >= 0) && (vy < RES) &&
                       (vz >= 0) && (vz < RES);
    if (!valid) return;

    const float cx = ((float)vx + 0.5f) * TRUNC - 0.5f;
    const float cy = ((float)vy + 0.5f) * TRUNC - 0.5f;
    const float cz = ((float)vz + 0.5f) * TRUNC - 0.5f;
    const float dx = x - cx, dy = y - cy, dz = z - cz;

    float dist = sqrtf(fmaf(dx, dx, fmaf(dy, dy, fmaf(dz, dz, 1e-12f))));
    dist = fminf(dist, TRUNC);

    const size_t lin = ((((size_t)b * RES + vx) * RES + vy) * RES) + vz;

    // count += 1 (bits 63:42), sum += dist (fixed point 2^-30, bits 41:0).
    // No overflow even if every pixel of an image hits one voxel:
    // 230400 * 2^23 < 2^42, and count 230400 < 2^22.
    const unsigned long long inc =
        (unsigned long long)CNT_ONE |
        (unsigned long long)(dist * SUM_SCALE + 0.5f);

    atomicAdd(acc + lin, inc);     // -> global_atomic_add_u64
}

// ---- Phase 3: finalize (4 voxels/thread, NT b128 loads, b128 store) -------
__global__ void __launch_bounds__(256)
finalize_kernel(const u64x2* __restrict__ acc2, float* __restrict__ out) {
    const size_t i = ((size_t)blockIdx.x * blockDim.x + threadIdx.x) * 2;

    // Accumulator is single-use here: nontemporal (TH=NT) 128-bit loads.
    const u64x2 a01 = __builtin_nontemporal_load(acc2 + i + 0);
    const u64x2 a23 = __builtin_nontemporal_load(acc2 + i + 1);

    unsigned long long a[4] = {a01.x, a01.y, a23.x, a23.y};

    float4 r;
    #pragma unroll
    for (int k = 0; k < 4; ++k) {
        const unsigned long long cnt = a[k] >> 42;
        const unsigned long long sf  = a[k] & SUM_MASK;
        float o = 0.0f;                               // empty voxel: 1 - 128*trunc = 0
        if (cnt != 0ull) {
            const float tdf = ((float)sf * INV_SUM_SCALE) / (float)cnt;
            o = 1.0f - (float)RES * tdf;
        }
        (&r.x)[k] = o;
    }
    *reinterpret_cast<float4*>(out + i * 2) = r;      // global_store_b128
}

// ---------------------------------------------------------------------------
extern "C" void kernel_launch(void* const* d_in, const int* in_sizes, int n_in,
                              void* d_out, int out_size, void* d_ws, size_t ws_size,
                              hipStream_t stream) {
    const float* depth    = (const float*)d_in[0];   // (16,1,480,480) fp32
    const float* fl       = (const float*)d_in[1];   // (16,1) fp32
    const float* cam_dist = (const float*)d_in[2];   // (16,1) fp32
    float*       out      = (float*)d_out;           // (16,1,128,128,128) fp32

    const int B        = in_sizes[1];                // 16
    const size_t n_vox = (size_t)out_size;           // 16 * 128^3 = 33,554,432

    unsigned long long* acc = (unsigned long long*)d_ws;  // n_vox * 8 bytes

    // Phase 1: clear accumulators (4 u64 = 32 B per thread)
    const int clear_blocks = (int)(n_vox / (256 * 4));
    clear_acc_kernel<<<dim3(clear_blocks), dim3(256), 0, stream>>>(
        reinterpret_cast<u64x2*>(acc));

    // Phase 2: scatter (one pixel per thread; 480*480 divisible by 256)
    dim3 grid_bp(NPIX / 256, B);
    backproject_kernel<<<grid_bp, dim3(256), 0, stream>>>(depth, fl, cam_dist, acc);

    // Phase 3: finalize (4 voxels per thread)
    const int fin_blocks = (int)(n_vox / (256 * 4));
    finalize_kernel<<<dim3(fin_blocks), dim3(256), 0, stream>>>(
        reinterpret_cast<const u64x2*>(acc), out);
}